// SparsePrototypeAlignment_23424751633122
// MI455X (gfx1250) — compile-verified
//
#include <hip/hip_runtime.h>
#include <hip/hip_bf16.h>
#include <math.h>
#include <stdint.h>

// Problem constants (from reference)
#define BSZ   64
#define CCH   2048
#define HW    288
#define KREG  8
#define KSHOT 4
#define NCLS  500
#define FDIM  (CCH * KREG)   // 16384

typedef float v2f __attribute__((ext_vector_type(2)));
typedef float v8f __attribute__((ext_vector_type(8)));

// ---------------------------------------------------------------------------
// PCG hash -> Box-Muller normal (stand-in for jax.random.normal(key 42)).
// ---------------------------------------------------------------------------
__device__ __forceinline__ unsigned int pcg_hash(unsigned int s) {
    s = s * 747796405u + 2891336453u;
    unsigned int w = ((s >> ((s >> 28u) + 4u)) ^ s) * 277803737u;
    return (w >> 22u) ^ w;
}

__device__ __forceinline__ float hash_normal(unsigned int i) {
    unsigned int a = pcg_hash(i);
    unsigned int b = pcg_hash(i ^ 0x9E3779B9u);
    float u1 = ((float)a + 1.0f) * 2.3283064365386963e-10f;  // (0,1]
    float u2 = (float)b * 2.3283064365386963e-10f;           // [0,1)
    return sqrtf(-2.0f * __logf(u1)) * __cosf(6.28318530718f * u2);
}

// ---------------------------------------------------------------------------
// Kernel 1: top-8 of cam per batch row. One wave (32 lanes) per batch item.
// Repeated argmax; tie-break = lowest index (matches jax.lax.top_k).
// ---------------------------------------------------------------------------
__global__ void topk_kernel(const float* __restrict__ cam, int* __restrict__ idx_out) {
    __shared__ float vals[HW];
    const int b = blockIdx.x;
    const int lane = threadIdx.x;
    for (int i = lane; i < HW; i += 32) vals[i] = cam[b * HW + i];
    __syncthreads();
    for (int t = 0; t < KREG; ++t) {
        float bv = -1e30f;
        int   bi = 0x7fffffff;
        for (int i = lane; i < HW; i += 32) {
            float v = vals[i];
            if (v > bv || (v == bv && i < bi)) { bv = v; bi = i; }
        }
        for (int m = 16; m >= 1; m >>= 1) {
            float ov = __shfl_xor(bv, m, 32);
            int   oi = __shfl_xor(bi, m, 32);
            if (ov > bv || (ov == bv && oi < bi)) { bv = ov; bi = oi; }
        }
        if (lane == 0) {
            idx_out[b * KREG + t] = bi;
            vals[bi] = -1e30f;
        }
        __syncthreads();
    }
}

// ---------------------------------------------------------------------------
// Kernel 2: rank/keep per sample + per-class kept-count and total-count.
// ---------------------------------------------------------------------------
__global__ void stats_kernel(const int* __restrict__ labels,
                             float* __restrict__ keep,
                             float* __restrict__ cnt_kept,
                             float* __restrict__ total) {
    __shared__ int   slab[BSZ];
    __shared__ float skeep[BSZ];
    const int t = threadIdx.x;
    if (t < BSZ) slab[t] = labels[t];
    __syncthreads();
    if (t < BSZ) {
        const int L = slab[t];
        int r = 0;
        for (int b = 0; b < t; ++b) r += (slab[b] == L) ? 1 : 0;
        const float k = (r < KSHOT) ? 1.0f : 0.0f;
        skeep[t] = k;
        keep[t]  = k;
    }
    __syncthreads();
    if (t < NCLS) {
        float ck = 0.0f, tt = 0.0f;
        for (int b = 0; b < BSZ; ++b) {
            if (slab[b] == t) { tt += 1.0f; ck += skeep[b]; }
        }
        cnt_kept[t] = ck;
        total[t]    = tt;
    }
}

// ---------------------------------------------------------------------------
// Kernel 3: gather top-k feature columns -> features (BSZ x FDIM).
// ---------------------------------------------------------------------------
__global__ void gather_kernel(const float* __restrict__ fm,
                              const int* __restrict__ idx,
                              float* __restrict__ feats) {
    __shared__ int sidx[KREG];
    const int b = blockIdx.y;
    const int t = blockIdx.x * blockDim.x + threadIdx.x;   // [0, FDIM)
    if (threadIdx.x < KREG) sidx[threadIdx.x] = idx[b * KREG + threadIdx.x];
    __syncthreads();
    const int c = t >> 3;
    const int j = t & 7;
    feats[(size_t)b * FDIM + t] =
        fm[(size_t)b * CCH * HW + (size_t)c * HW + sidx[j]];
}

// ---------------------------------------------------------------------------
// Kernel 4: sums = onehot^T @ (features * keep) via V_WMMA_F32_16X16X4_F32.
// One wave per block; each block computes FOUR 16x16 class tiles sharing one
// 64x16 feature B-tile that is DMA-staged into LDS with
// GLOBAL_LOAD_ASYNC_TO_LDS_B128 (ASYNCcnt path), overlapped with building
// branchless keep-masked one-hot bitmasks for the A operand.
// grid = (FDIM/16, 8), block = 32.
//
// A (16x4 f32, 2 VGPRs): lane l holds M = l%16, K = kb + 2*(l/16) + {0,1}
// B (4x16 f32, 2 VGPRs): lane l holds N = l%16, K = kb + 2*(l/16) + {0,1}
// D (16x16 f32, 8 VGPRs): lane l, reg r -> (M = r + 8*(l/16), N = l%16)
// ---------------------------------------------------------------------------
__global__ void proto_wmma_kernel(const float* __restrict__ feats,
                                  const int*   __restrict__ labels,
                                  const float* __restrict__ keep,
                                  const float* __restrict__ cnt_kept,
                                  const float* __restrict__ total,
                                  const float* __restrict__ prototypes,
                                  const float* __restrict__ counts,
                                  const int*   __restrict__ modality_p,
                                  float* __restrict__ out) {
    __shared__ float btile[BSZ * 16];   // 4 KB: feats[0..63][n_base..n_base+15]
    __shared__ int   slab[BSZ];
    __shared__ float skeep[BSZ];

    const int lane   = threadIdx.x;          // wave32
    const int n_base = blockIdx.x * 16;
    const int cls0   = blockIdx.y * 64;      // 4 class tiles of 16

    for (int i = lane; i < BSZ; i += 32) {
        slab[i]  = labels[i];
        skeep[i] = keep[i];
    }

    // ---- async DMA: stage 64x16 f32 feature tile into LDS (8 x b128/lane) --
    const unsigned int ldsbase = (unsigned int)(uintptr_t)btile; // low 32b = LDS offset
    {
        const int rq = lane >> 2;            // row sub-id 0..7
        const int q  = (lane & 3) * 4;       // float column group
        #pragma unroll
        for (int i = 0; i < 8; ++i) {
            const int b = i * 8 + rq;
            const float* gp = feats + (size_t)b * FDIM + n_base + q;
            const unsigned int la = ldsbase + (unsigned int)(b * 16 + q) * 4u;
            asm volatile("global_load_async_to_lds_b128 %0, %1, off"
                         :: "v"(la), "v"(gp) : "memory");
        }
    }
    __syncthreads();                          // slab/skeep visible

    const int half = lane >> 4;               // 0: lanes 0-15, 1: lanes 16-31
    const int m    = lane & 15;
    const int n    = n_base + m;
    const int mod  = modality_p[0];

    // warm L2 for the prototype rows this block's epilogue will read
    __builtin_prefetch(&prototypes[((size_t)(cls0 + m) * 2 + mod) * FDIM + n], 0, 1);

    // ---- branchless keep-masked one-hot: one 64-bit mask per class tile ----
    unsigned long long wmask[4];
    #pragma unroll
    for (int t = 0; t < 4; ++t) {
        const int myc = cls0 + t * 16 + m;
        unsigned long long mk = 0ull;
        for (int b = 0; b < BSZ; ++b) {
            const unsigned long long hit =
                (unsigned long long)((slab[b] == myc) & (skeep[b] > 0.0f));
            mk |= hit << b;
        }
        wmask[t] = mk;
    }

    asm volatile("s_wait_asynccnt 0" ::: "memory");  // B tile resident in LDS
    __syncthreads();

    v8f accs[4] = {{}, {}, {}, {}};
    #pragma unroll
    for (int kb = 0; kb < BSZ; kb += 4) {
        const int ka = kb + 2 * half;
        v2f bb;
        bb.x = btile[ka * 16 + m];
        bb.y = btile[(ka + 1) * 16 + m];
        #pragma unroll
        for (int t = 0; t < 4; ++t) {
            v2f a;
            a.x = (float)((wmask[t] >> ka) & 1ull);
            a.y = (float)((wmask[t] >> (ka + 1)) & 1ull);
            accs[t] = __builtin_amdgcn_wmma_f32_16x16x4_f32(
                false, a, false, bb, (short)0, accs[t], false, false);
        }
    }

    // ---- epilogue: mean -> select(rand_init/old) -> EMA -------------------
    #pragma unroll
    for (int t = 0; t < 4; ++t) {
        #pragma unroll
        for (int r = 0; r < 8; ++r) {
            const int cls = cls0 + t * 16 + r + 8 * half;
            if (cls >= NCLS) continue;
            const float ck   = cnt_kept[cls];
            const float tt   = total[cls];
            const float cst  = counts[cls * 2 + mod];
            const float oldv = prototypes[((size_t)cls * 2 + mod) * FDIM + n];
            const float mean = accs[t][r] / fmaxf(ck, 1.0f);
            const float ri   = 0.01f * hash_normal((unsigned int)(cls * FDIM + n));
            const float bp   = (tt > 0.0f) ? mean : ((cst == 0.0f) ? ri : oldv);
            out[(size_t)cls * FDIM + n] = 0.9f * oldv + 0.1f * bp;
        }
    }
}

// ---------------------------------------------------------------------------
// Kernel 5: fused row L2-normalize. Block-per-row; the 64 KB row is staged in
// LDS (minus a small tail to stay under 64 KB static LDS) so the row is read
// from HBM exactly once and written exactly once.
// ---------------------------------------------------------------------------
#define STAGE (FDIM - 32)
__global__ void norm_scale_kernel(float* __restrict__ out) {
    __shared__ float row[STAGE];
    __shared__ float red[8];
    __shared__ float sinv;
    const int r = blockIdx.x;
    float ss = 0.0f;
    for (int i = threadIdx.x; i < FDIM; i += blockDim.x) {
        const float v = out[(size_t)r * FDIM + i];
        if (i < STAGE) row[i] = v;
        ss += v * v;
    }
    for (int m = 16; m >= 1; m >>= 1) ss += __shfl_xor(ss, m, 32);
    if ((threadIdx.x & 31) == 0) red[threadIdx.x >> 5] = ss;
    __syncthreads();
    if (threadIdx.x == 0) {
        float s = 0.0f;
        for (int i = 0; i < 8; ++i) s += red[i];
        sinv = 1.0f / (sqrtf(s) + 1e-8f);
    }
    __syncthreads();
    const float inv = sinv;
    for (int i = threadIdx.x; i < FDIM; i += blockDim.x) {
        const float v = (i < STAGE) ? row[i] : out[(size_t)r * FDIM + i];
        out[(size_t)r * FDIM + i] = v * inv;
    }
}

// ---------------------------------------------------------------------------
extern "C" void kernel_launch(void* const* d_in, const int* in_sizes, int n_in,
                              void* d_out, int out_size, void* d_ws, size_t ws_size,
                              hipStream_t stream) {
    const float* fm       = (const float*)d_in[0];  // feature_map (64,2048,24,12)
    const float* cam      = (const float*)d_in[1];  // cam (64,24,12)
    const int*   labels   = (const int*)  d_in[2];  // labels (64,)
    const float* protos   = (const float*)d_in[3];  // prototypes (500,2,16384)
    const float* counts   = (const float*)d_in[4];  // counts (500,2)
    const int*   modality = (const int*)  d_in[5];  // scalar
    float*       out      = (float*)d_out;          // (500,16384)

    // Workspace layout (~4.2 MB total)
    char*  ws       = (char*)d_ws;
    float* feats    = (float*)ws;                               // 64*16384*4 = 4 MB
    int*   idx      = (int*)  (ws + (size_t)BSZ * FDIM * 4);    // 64*8 ints
    float* keep     = (float*)(ws + (size_t)BSZ * FDIM * 4 + 2048);
    float* cnt_kept = keep + 64;
    float* total    = cnt_kept + 512;

    topk_kernel  <<<BSZ, 32, 0, stream>>>(cam, idx);
    stats_kernel <<<1, 512, 0, stream>>>(labels, keep, cnt_kept, total);
    gather_kernel<<<dim3(FDIM / 256, BSZ), 256, 0, stream>>>(fm, idx, feats);
    proto_wmma_kernel<<<dim3(FDIM / 16, 8), 32, 0, stream>>>(
        feats, labels, keep, cnt_kept, total, protos, counts, modality, out);
    norm_scale_kernel<<<NCLS, 256, 0, stream>>>(out);
}